// IIREnvelopeFollower_10694468567162
// MI455X (gfx1250) — compile-verified
//
#include <hip/hip_runtime.h>
#include <hip/hip_bf16.h>
#include <math.h>

// Problem constants (from reference)
#define B_    32
#define T_    131072
#define KLEN  16384
#define SEG   2048          // segment length handled by one block in pass 2/4
#define NSEG  (T_ / SEG)    // 64 segments per batch row

typedef __attribute__((ext_vector_type(2))) float v2f;
typedef __attribute__((ext_vector_type(8))) float v8f;

__device__ __forceinline__ float alpha_of(float z) {
    float sig = 1.0f / (1.0f + __expf(-z));
    return fminf(fmaxf(sig, 1e-5f), 1.0f - 1e-5f);
}

// ---------------------------------------------------------------------------
// Pass 1: energy[b][t] = 0.5*(s[b][0][t]^2 + s[b][1][t]^2)   (float4 streams)
// ---------------------------------------------------------------------------
__global__ void __launch_bounds__(256) k_energy(const float* __restrict__ sig,
                                                float* __restrict__ e) {
    int gid  = blockIdx.x * 256 + threadIdx.x;
    int flat = gid * 4;                  // B_*T_ = 4,194,304 elements total
    int b = flat / T_;
    int t = flat % T_;
    const float4 s0 = *(const float4*)(sig + (size_t)b * 2 * T_ + t);
    const float4 s1 = *(const float4*)(sig + (size_t)b * 2 * T_ + T_ + t);
    float4 r;
    r.x = 0.5f * (s0.x * s0.x + s1.x * s1.x);
    r.y = 0.5f * (s0.y * s0.y + s1.y * s1.y);
    r.z = 0.5f * (s0.z * s0.z + s1.z * s1.z);
    r.w = 0.5f * (s0.w * s0.w + s1.w * s1.w);
    *(float4*)(e + flat) = r;
}

// ---------------------------------------------------------------------------
// Pass 2: segment-local end value  P[b][s] = sum_i alpha^(SEG-1-i) * b[t0+i]
//         with b[t] = (1-a)*(e[t] - a^K * e[t-K])
// ---------------------------------------------------------------------------
__global__ void __launch_bounds__(256) k_partial(const float* __restrict__ e,
                                                 const float* __restrict__ zalpha,
                                                 float* __restrict__ P) {
    const int s = blockIdx.x, b = blockIdx.y;
    const int j = threadIdx.x;                       // 0..255, 8 elems each
    const float alpha = alpha_of(zalpha[b]);
    const float la  = __logf(alpha);
    const float oma = 1.0f - alpha;
    const float aK  = __expf((float)KLEN * la);
    const float* er = e + (size_t)b * T_;
    const int t0 = s * SEG + j * 8;

    float p = 0.0f;                                   // Horner: Σ a^(7-i) b_i
#pragma unroll
    for (int i = 0; i < 8; ++i) {
        int t = t0 + i;
        float ek = (t >= KLEN) ? er[t - KLEN] : 0.0f;
        float bv = oma * (er[t] - aK * ek);
        p = fmaf(alpha, p, bv);
    }

    __shared__ float red[256];
    // weight so the block-sum equals the segment-local end value
    red[j] = p * __expf(8.0f * (float)(255 - j) * la);
    __syncthreads();
#pragma unroll
    for (int off = 128; off > 0; off >>= 1) {
        if (j < off) red[j] += red[j + off];
        __syncthreads();
    }
    if (j == 0) P[b * NSEG + s] = red[0];
}

// ---------------------------------------------------------------------------
// Pass 3: tiny per-batch scan of segment carries (one lane per batch row)
// ---------------------------------------------------------------------------
__global__ void k_scan(const float* __restrict__ P,
                       const float* __restrict__ zalpha,
                       float* __restrict__ Carr) {
    int b = threadIdx.x;
    if (b < B_) {
        float alpha = alpha_of(zalpha[b]);
        float aL = __expf((float)SEG * __logf(alpha));
        float c = 0.0f;
        for (int s = 0; s < NSEG; ++s) {
            Carr[b * NSEG + s] = c;          // carry INTO segment s
            c = fmaf(aL, c, P[b * NSEG + s]);
        }
    }
}

// ---------------------------------------------------------------------------
// Pass 4: per segment — WMMA triangular scan + carry combine + relu/log
//   Each wave owns a 256-step tile = 16 chunks x 16 steps.
//   Y_local(16x16) = L(16x16, lower-tri decay) @ E(16x16, chunks as columns)
//   done with 4x V_WMMA_F32_16X16X4_F32 (K sliced 4 at a time).
// ---------------------------------------------------------------------------
__global__ void __launch_bounds__(256) k_final(const float* __restrict__ e,
                                               const float* __restrict__ zalpha,
                                               const float* __restrict__ Carr,
                                               float* __restrict__ out) {
    const int s = blockIdx.x, b = blockIdx.y;
    const int tid  = threadIdx.x;
    const int wave = tid >> 5;
    const int lane = tid & 31;

    __shared__ float lbs[8][256];   // per-wave tile staging (b-values, then output)
    __shared__ float cend[8][16];   // per-wave chunk-end local values
    __shared__ float wend[8];       // per-wave tile-end local values

    const float alpha = alpha_of(zalpha[b]);
    const float la   = __logf(alpha);
    const float oma  = 1.0f - alpha;
    const float aK   = __expf((float)KLEN * la);
    const float a16  = __expf(16.0f * la);
    const float* er  = e + (size_t)b * T_;
    const int segStart  = s * SEG;
    const int tileStart = segStart + wave * 256;

    // ---- stage 1: compute b[t] for this wave's 256 steps into LDS (coalesced)
    {
        const int i0 = lane * 8;
#pragma unroll
        for (int i = 0; i < 8; ++i) {
            int t = tileStart + i0 + i;
            float ek = (t >= KLEN) ? er[t - KLEN] : 0.0f;
            lbs[wave][i0 + i] = oma * (er[t] - aK * ek);
        }
    }
    __syncthreads();

    // ---- stage 2: Y_local = L @ E via 4x WMMA f32 16x16x4
    const int half = lane >> 4;      // 0: K-slots {0,1}; 1: K-slots {2,3}
    const int m    = lane & 15;      // A row index == B column (chunk) index
    v8f c = {};
#pragma unroll
    for (int sl = 0; sl < 4; ++sl) {
        const int k0 = 4 * sl + 2 * half;
        v2f a;                        // A[m][k0], A[m][k0+1] : lower-tri decay
        a.x = (m >= k0)     ? __expf((float)(m - k0) * la)     : 0.0f;
        a.y = (m >= k0 + 1) ? __expf((float)(m - k0 - 1) * la) : 0.0f;
        v2f bb;                       // E[k0][m], E[k0+1][m]  (chunk m column)
        bb.x = lbs[wave][16 * m + k0];
        bb.y = lbs[wave][16 * m + k0 + 1];
        c = __builtin_amdgcn_wmma_f32_16x16x4_f32(
                /*neg_a=*/false, a, /*neg_b=*/false, bb,
                /*c_mod=*/(short)0, c, /*reuse_a=*/false, /*reuse_b=*/false);
    }

    // ---- stage 3: carry hierarchy
    // chunk-end local values: row M=15 lives in VGPR 7 of lanes 16..31
    if (half == 1) cend[wave][m] = c[7];
    __syncthreads();

    float myCarry = 0.0f, cc = 0.0f;   // scan 16 chunk ends (affine compose)
#pragma unroll
    for (int j = 0; j < 16; ++j) {
        if (j == m) myCarry = cc;
        cc = fmaf(a16, cc, cend[wave][j]);
    }
    if (lane == 0) wend[wave] = cc;    // tile-end local value
    __syncthreads();

    float cw = Carr[b * NSEG + s] * __expf((float)(256 * wave) * la);
    for (int v = 0; v < wave; ++v)
        cw = fmaf(wend[v], __expf((float)(256 * (wave - 1 - v)) * la), cw);
    const float carryChunk = fmaf(cw, __expf((float)(16 * m) * la), myCarry);

    // ---- stage 4: y = local + carry*alpha^(m+1); relu; log; stage via LDS
#pragma unroll
    for (int r = 0; r < 8; ++r) {
        const int mm = r + 8 * half;
        float val = fmaf(carryChunk, __expf((float)(mm + 1) * la), c[r]);
        float y = fmaxf(val, 0.0f);
        lbs[wave][16 * m + mm] = __logf(y + 1e-5f);
    }
    __syncthreads();

    // coalesced float4 stores
    float* orow = out + (size_t)b * T_ + tileStart;
    const int i0 = lane * 8;
    float4 o0, o1;
    o0.x = lbs[wave][i0 + 0]; o0.y = lbs[wave][i0 + 1];
    o0.z = lbs[wave][i0 + 2]; o0.w = lbs[wave][i0 + 3];
    o1.x = lbs[wave][i0 + 4]; o1.y = lbs[wave][i0 + 5];
    o1.z = lbs[wave][i0 + 6]; o1.w = lbs[wave][i0 + 7];
    *(float4*)(orow + i0)     = o0;
    *(float4*)(orow + i0 + 4) = o1;
}

// ---------------------------------------------------------------------------
extern "C" void kernel_launch(void* const* d_in, const int* in_sizes, int n_in,
                              void* d_out, int out_size, void* d_ws, size_t ws_size,
                              hipStream_t stream) {
    const float* signal = (const float*)d_in[0];   // (32, 2, 131072) f32
    const float* zalpha = (const float*)d_in[1];   // (32, 1) f32
    float* out = (float*)d_out;                    // (32, 131072) f32

    float* e    = (float*)d_ws;                    // B_*T_ floats (16.78 MB)
    float* P    = e + (size_t)B_ * T_;             // B_*NSEG floats
    float* Carr = P + (size_t)B_ * NSEG;           // B_*NSEG floats

    k_energy <<<dim3((B_ * T_) / (256 * 4)), 256, 0, stream>>>(signal, e);
    k_partial<<<dim3(NSEG, B_),             256, 0, stream>>>(e, zalpha, P);
    k_scan   <<<dim3(1),                     32, 0, stream>>>(P, zalpha, Carr);
    k_final  <<<dim3(NSEG, B_),             256, 0, stream>>>(e, zalpha, Carr, out);
}